// Couple_loss_62380105007762
// MI455X (gfx1250) — compile-verified
//
#include <hip/hip_runtime.h>
#include <hip/hip_bf16.h>
#include <stdint.h>

// Problem constants (fixed by setup_inputs).
#define BB      64
#define ZZ      256
#define FF      128
#define TT      2048
#define CC      5
#define TILE_T  128                 // t-columns per block (4 waves x 32 columns)
#define NTILES  (TT / TILE_T)       // 16
#define NBLOCKS (BB * NTILES)       // 1024
#define RSTR    48                  // padded row stride (floats): rows land on disjoint bank quadrants

typedef __attribute__((ext_vector_type(2))) float v2f;
typedef __attribute__((ext_vector_type(8))) float v8f;

// Per-block partial of sum_{t in tile} (sum_f diff[f][t])^2, channel 0 only.
//
// Data path: HBM -> LDS via global_load_async_to_lds_b128 (double-buffered,
// per-wave private sub-tiles, so the only inner-loop sync is the wave's own
// ASYNCcnt -- no workgroup barriers). F-reduction runs on the matrix pipe:
// D = ones(16x4) x B(4x16) + C via v_wmma_f32_16x16x4_f32, f32 end to end.
__global__ void __launch_bounds__(TILE_T) mse_partial_kernel(
    const float* __restrict__ out_rec,
    const float* __restrict__ tgt_rec,
    float* __restrict__ partials) {
  // [buffer][tensor(out/tgt)][wave][row][col(padded)]
  __shared__ __align__(16) float sbuf[2][2][4][4][RSTR];
  __shared__ float red[TILE_T];

  const int tid  = threadIdx.x;            // 0..127
  const int b    = blockIdx.x >> 4;        // / NTILES
  const int tile = blockIdx.x & (NTILES - 1);

  const int wave = tid >> 5;               // 0..3
  const int lane = tid & 31;

  // staging role: lane -> (row 0..3, 4-float column group) of the wave's 4x32 sub-tile
  const int srow  = lane >> 3;             // 0..3
  const int scol4 = (lane & 7) << 2;       // 0,4,...,28

  // consume role
  const int clA    = lane & 15;            // local column, group A
  const int clB    = clA + 16;             // local column, group B
  const int row_lo = (lane >> 4) << 1;     // lanes 0-15 -> rows {0,1}, 16-31 -> {2,3}

  // channel 0, this block's tile, this wave's 32-column slice
  const size_t gbase = (size_t)b * (2 * FF * TT) + (size_t)tile * TILE_T + (size_t)(wave << 5);

  v2f ones; ones.x = 1.0f; ones.y = 1.0f;
  v8f acc0 = {};
  v8f acc1 = {};

  // issue one async HBM->LDS pair (out + tgt sub-tile rows) for chunk fc into buffer bufi
  #define ISSUE_CHUNK(bufi, fc)                                                        \
    do {                                                                               \
      const size_t goff = gbase + (size_t)(4 * (fc) + srow) * TT + scol4;              \
      const uint64_t go = (uint64_t)(uintptr_t)(out_rec + goff);                       \
      const uint64_t gt = (uint64_t)(uintptr_t)(tgt_rec + goff);                       \
      const uint32_t lo = (uint32_t)(uintptr_t)&sbuf[(bufi)][0][wave][srow][scol4];    \
      const uint32_t lt = (uint32_t)(uintptr_t)&sbuf[(bufi)][1][wave][srow][scol4];    \
      asm volatile("global_load_async_to_lds_b128 %0, %1, off"                         \
                   :: "v"(lo), "v"(go) : "memory");                                    \
      asm volatile("global_load_async_to_lds_b128 %0, %1, off"                         \
                   :: "v"(lt), "v"(gt) : "memory");                                    \
    } while (0)

  ISSUE_CHUNK(0, 0);

  for (int fc = 0; fc < FF / 4; ++fc) {
    const int cb = fc & 1;
    // prefetch next chunk into the other buffer ((fc+1)&31 wraps: harmless dummy on last iter)
    ISSUE_CHUNK(cb ^ 1, (fc + 1) & 31);
    // in-order async completion: <=2 outstanding means current buffer's pair is done
    asm volatile("s_wait_asynccnt 0x2" ::: "memory");

    v2f bA, bB;
    bA.x = sbuf[cb][0][wave][row_lo    ][clA] - sbuf[cb][1][wave][row_lo    ][clA];
    bA.y = sbuf[cb][0][wave][row_lo + 1][clA] - sbuf[cb][1][wave][row_lo + 1][clA];
    bB.x = sbuf[cb][0][wave][row_lo    ][clB] - sbuf[cb][1][wave][row_lo    ][clB];
    bB.y = sbuf[cb][0][wave][row_lo + 1][clB] - sbuf[cb][1][wave][row_lo + 1][clB];

    acc0 = __builtin_amdgcn_wmma_f32_16x16x4_f32(false, ones, false, bA,
                                                 (short)0, acc0, false, false);
    acc1 = __builtin_amdgcn_wmma_f32_16x16x4_f32(false, ones, false, bB,
                                                 (short)0, acc1, false, false);
  }
  // drain the trailing dummy issue before LDS reuse / wave end
  asm volatile("s_wait_asynccnt 0x0" ::: "memory");
  #undef ISSUE_CHUNK

  // Every row of D holds the column sums; lanes 16-31 duplicate lanes 0-15.
  float s0 = acc0[0];
  float s1 = acc1[0];
  float p  = (lane < 16) ? (s0 * s0 + s1 * s1) : 0.0f;

  red[tid] = p;
  __syncthreads();
  for (int s = TILE_T / 2; s > 0; s >>= 1) {
    if (tid < s) red[tid] += red[tid + s];
    __syncthreads();
  }
  if (tid == 0) partials[blockIdx.x] = red[0];
}

// Single block: combine MSE partials + KLD + cross-entropy, fixed-order reduction.
__global__ void __launch_bounds__(256) finalize_kernel(
    const float* __restrict__ mean,
    const float* __restrict__ log_var,
    const float* __restrict__ out_clas,
    const int*   __restrict__ tgt_clas,
    const float* __restrict__ weight,
    const float* __restrict__ partials,
    float* __restrict__ out) {
  __shared__ float red[256];
  const int tid = threadIdx.x;

  const float w0 = weight[0], w1 = weight[1], w2 = weight[2];

  // MSE partial sums (1024 slots)
  float mse = 0.0f;
  for (int i = tid; i < NBLOCKS; i += 256) mse += partials[i];

  // KLD raw sum over B*Z = 16384 elements: sum(1 + lv - m^2 - exp(lv))
  float kldsum = 0.0f;
  for (int i = tid; i < BB * ZZ; i += 256) {
    float m  = mean[i];
    float lv = log_var[i];
    kldsum += 1.0f + lv - m * m - expf(lv);
  }

  // Cross-entropy: threads 0..63 each handle one row of [B, C]
  float ce = 0.0f;
  if (tid < BB) {
    const float* x = out_clas + tid * CC;
    float mx = x[0];
    #pragma unroll
    for (int c = 1; c < CC; ++c) mx = fmaxf(mx, x[c]);
    float se = 0.0f;
    #pragma unroll
    for (int c = 0; c < CC; ++c) se += expf(x[c] - mx);
    const int t = tgt_clas[tid];
    ce = (mx + logf(se)) - x[t];
  }

  // final = w0 * 4*mse_raw + w1 * (-0.5)*kldsum + w2 * mean(ce)  (all linear -> one reduction)
  red[tid] = w0 * 4.0f * mse + w1 * (-0.5f) * kldsum + w2 * (1.0f / (float)BB) * ce;
  __syncthreads();
  for (int s = 128; s > 0; s >>= 1) {
    if (tid < s) red[tid] += red[tid + s];
    __syncthreads();
  }
  if (tid == 0) out[0] = red[0];
}

extern "C" void kernel_launch(void* const* d_in, const int* in_sizes, int n_in,
                              void* d_out, int out_size, void* d_ws, size_t ws_size,
                              hipStream_t stream) {
  const float* mean     = (const float*)d_in[0];   // [64,256]
  const float* log_var  = (const float*)d_in[1];   // [64,256]
  const float* out_rec  = (const float*)d_in[2];   // [64,2,128,2048]
  const float* out_clas = (const float*)d_in[3];   // [64,5]
  const float* tgt_rec  = (const float*)d_in[4];   // [64,2,128,2048]
  const int*   tgt_clas = (const int*)d_in[5];     // [64]
  // d_in[6] = batch_size (64), d_in[7] = img_size (2048): fixed, unused
  const float* weight   = (const float*)d_in[8];   // [3]

  float* partials = (float*)d_ws;                  // 1024 floats, every slot written each call

  mse_partial_kernel<<<NBLOCKS, TILE_T, 0, stream>>>(out_rec, tgt_rec, partials);
  finalize_kernel<<<1, 256, 0, stream>>>(mean, log_var, out_clas, tgt_clas,
                                         weight, partials, (float*)d_out);
}